// GCN_53163105190492
// MI455X (gfx1250) — compile-verified
//
#include <hip/hip_runtime.h>

// GCN on MI455X (gfx1250, wave32).
// Pipeline: deg -> dinv -> [GEMM(wmma f32, LDS-staged B) -> copy(selfloop) -> scatter-atomic -> post]x2 -> GEMM(out)
// ws layout (floats): dinv[N] | h[N*64] | a[N*64]  (~51.6 MB)

typedef __attribute__((ext_vector_type(2))) float v2f;
typedef __attribute__((ext_vector_type(8))) float v8f;

#define FEAT 64

__global__ void k_fill(float* __restrict__ p, float v, int n) {
    int i = blockIdx.x * blockDim.x + threadIdx.x;
    if (i < n) p[i] = v;
}

__global__ void k_deg(const int* __restrict__ dst, float* __restrict__ deg, int nE) {
    int e = blockIdx.x * blockDim.x + threadIdx.x;
    if (e < nE) atomicAdd(&deg[dst[e]], 1.0f);
}

__global__ void k_rsqrt(float* __restrict__ d, int n) {
    int i = blockIdx.x * blockDim.x + threadIdx.x;
    if (i < n) d[i] = rsqrtf(d[i]);
}

__global__ void k_copy4(const float4* __restrict__ s, float4* __restrict__ d, int n4) {
    int i = blockIdx.x * blockDim.x + threadIdx.x;
    if (i < n4) d[i] = s[i];
}

// 16 lanes per edge, each lane moves a float4 chunk of the 64-wide feature row.
// agg was pre-initialized with the self-loop term; real edges accumulate via L2 atomics
// (h and agg are 25.6 MB each -> both L2-resident on the 192 MB L2).
__global__ void k_scatter(const float* __restrict__ h, float* __restrict__ agg,
                          const int* __restrict__ src, const int* __restrict__ dst, int nE) {
    unsigned long long gid = (unsigned long long)blockIdx.x * blockDim.x + threadIdx.x;
    int e = (int)(gid >> 4);
    if (e >= nE) return;
    int c = ((int)gid & 15) << 2;
    int s = src[e], d = dst[e];
    const float4 v = *(const float4*)(h + (size_t)s * FEAT + c);
    float* p = agg + (size_t)d * FEAT + c;
    atomicAdd(p + 0, v.x);
    atomicAdd(p + 1, v.y);
    atomicAdd(p + 2, v.z);
    atomicAdd(p + 3, v.w);
}

// a[i] = (a[i] * dinv[row] + bias[col]) with optional relu
__global__ void k_post(float* __restrict__ a, const float* __restrict__ dinv,
                       const float* __restrict__ bias, int total, int do_relu) {
    int i = blockIdx.x * blockDim.x + threadIdx.x;
    if (i >= total) return;
    int row = i >> 6, c = i & 63;
    float v = a[i] * dinv[row] + bias[c];
    if (do_relu) v = fmaxf(v, 0.0f);
    a[i] = v;
}

// C[M x 64] = (A[M x K] @ B[K x 64]) * rowscale[row] + bias[col]
// One wave per 16-row strip, 4 N-tiles of 16x16 via V_WMMA_F32_16X16X4_F32 (full fp32).
// B is staged TRANSPOSED in LDS per K-panel: column pairs (B[kk][c], B[kk+1][c]) become
// contiguous -> one ds_load_b64 per tile per K-step. Stride S = KP+34: even (8B align for
// b64) and S % 64 == 34 -> the 32 lanes of a ds_load_b64 hit disjoint bank pairs.
// A-frag: lane loads its own 8B at (row0 + r, k + 2*half) -> no cndmask selects.
template <int K>
__global__ __launch_bounds__(128) void k_gemm_wmma(
    const float* __restrict__ A, const float* __restrict__ B, float* __restrict__ C,
    const float* __restrict__ rowscale, const float* __restrict__ bias, int M) {
    constexpr int KP = (K < 128) ? K : 128;     // K-panel staged in LDS
    constexpr int S  = KP + 34;                 // LDS column stride (floats)
    __shared__ float bs[FEAT * S];              // 41.5 KB for KP=128

    const int lane = threadIdx.x & 31;
    const int wave = threadIdx.x >> 5;
    const int strip = blockIdx.x * 4 + wave;
    const int row0 = strip << 4;
    const bool active = row0 < M;               // wave-uniform

    const int r = lane & 15;
    const int half = lane >> 4;
    const float* arow = active ? (A + (size_t)(row0 + r) * K) : A;
    const float* bcol = bs + r * S;

    v8f acc0{}, acc1{}, acc2{}, acc3{};

    for (int k0 = 0; k0 < K; k0 += KP) {
        __syncthreads();                        // previous panel fully consumed
        // cooperative transposed fill: coalesced float4 reads of W panel
        const float4* bp4 = (const float4*)(B + (size_t)k0 * FEAT);
        for (int idx = threadIdx.x; idx < KP * (FEAT / 4); idx += 128) {
            float4 w = bp4[idx];
            int kk = idx >> 4;                  // row within panel
            int c  = (idx & 15) << 2;           // starting column
            bs[(c + 0) * S + kk] = w.x;
            bs[(c + 1) * S + kk] = w.y;
            bs[(c + 2) * S + kk] = w.z;
            bs[(c + 3) * S + kk] = w.w;
        }
        __syncthreads();

        if (active) {
            for (int k = 0; k < KP; k += 4) {
                const int kk = k + half * 2;
                float2 a2 = *(const float2*)(arow + k0 + kk);
                v2f af;
                af.x = a2.x;
                af.y = a2.y;
                v2f b0 = *(const v2f*)(bcol + 0 * 16 * S + kk);
                v2f b1 = *(const v2f*)(bcol + 1 * 16 * S + kk);
                v2f b2 = *(const v2f*)(bcol + 2 * 16 * S + kk);
                v2f b3 = *(const v2f*)(bcol + 3 * 16 * S + kk);
                acc0 = __builtin_amdgcn_wmma_f32_16x16x4_f32(false, af, false, b0, (short)0, acc0, false, false);
                acc1 = __builtin_amdgcn_wmma_f32_16x16x4_f32(false, af, false, b1, (short)0, acc1, false, false);
                acc2 = __builtin_amdgcn_wmma_f32_16x16x4_f32(false, af, false, b2, (short)0, acc2, false, false);
                acc3 = __builtin_amdgcn_wmma_f32_16x16x4_f32(false, af, false, b3, (short)0, acc3, false, false);
            }
        }
    }

    if (!active) return;

    // D layout: VGPR i, lanes 0-15 -> row row0+i, lanes 16-31 -> row row0+i+8; col = tile*16 + r
    float sc[8];
#pragma unroll
    for (int i = 0; i < 8; ++i) {
        int row = row0 + i + half * 8;
        sc[i] = rowscale ? rowscale[row] : 1.0f;
    }
    float bb0 = 0.f, bb1 = 0.f, bb2 = 0.f, bb3 = 0.f;
    if (bias) { bb0 = bias[r]; bb1 = bias[16 + r]; bb2 = bias[32 + r]; bb3 = bias[48 + r]; }

#pragma unroll
    for (int i = 0; i < 8; ++i) {
        size_t off = (size_t)(row0 + i + half * 8) * FEAT + r;
        C[off +  0] = acc0[i] * sc[i] + bb0;
        C[off + 16] = acc1[i] * sc[i] + bb1;
        C[off + 32] = acc2[i] * sc[i] + bb2;
        C[off + 48] = acc3[i] * sc[i] + bb3;
    }
}

extern "C" void kernel_launch(void* const* d_in, const int* in_sizes, int n_in,
                              void* d_out, int out_size, void* d_ws, size_t ws_size,
                              hipStream_t stream) {
    const float* x    = (const float*)d_in[0];
    const int*   ei   = (const int*)d_in[1];
    const float* W1   = (const float*)d_in[2];
    const float* b1   = (const float*)d_in[3];
    const float* W2   = (const float*)d_in[4];
    const float* b2   = (const float*)d_in[5];
    const float* Wlin = (const float*)d_in[6];
    const float* blin = (const float*)d_in[7];
    float* out = (float*)d_out;

    const int K1 = 512;
    const int N  = in_sizes[0] / K1;   // 100000 (divisible by 16)
    const int E  = in_sizes[1] / 2;    // 3200000
    const int* src = ei;
    const int* dst = ei + E;

    float* ws   = (float*)d_ws;
    float* dinv = ws;                       // N
    float* h    = ws + N;                   // N*64
    float* a    = h + (size_t)N * FEAT;     // N*64

    const int T = 256;
    const int totalF = N * FEAT;
    const int n4 = totalF / 4;
    const int strips = N / 16;
    const int gblocks = (strips + 3) / 4;
    const long long sthreads = (long long)E * 16;
    const int sblocks = (int)((sthreads + T - 1) / T);

    // degree (self-loop -> init 1.0) and D^{-1/2}
    k_fill<<<(N + T - 1) / T, T, 0, stream>>>(dinv, 1.0f, N);
    k_deg<<<(E + T - 1) / T, T, 0, stream>>>(dst, dinv, E);
    k_rsqrt<<<(N + T - 1) / T, T, 0, stream>>>(dinv, N);

    // ---- layer 1 ----
    k_gemm_wmma<512><<<gblocks, 128, 0, stream>>>(x, W1, h, dinv, nullptr, N);    // h = (xW1)*dinv[row]
    k_copy4<<<(n4 + T - 1) / T, T, 0, stream>>>((const float4*)h, (float4*)a, n4); // self-loop term
    k_scatter<<<sblocks, T, 0, stream>>>(h, a, src, dst, E);
    k_post<<<(totalF + T - 1) / T, T, 0, stream>>>(a, dinv, b1, totalF, 1);       // *dinv[dst] + b1, relu

    // ---- layer 2 ----
    k_gemm_wmma<64><<<gblocks, 128, 0, stream>>>(a, W2, h, dinv, nullptr, N);
    k_copy4<<<(n4 + T - 1) / T, T, 0, stream>>>((const float4*)h, (float4*)a, n4);
    k_scatter<<<sblocks, T, 0, stream>>>(h, a, src, dst, E);
    k_post<<<(totalF + T - 1) / T, T, 0, stream>>>(a, dinv, b2, totalF, 0);       // no relu

    // ---- output linear ----
    k_gemm_wmma<64><<<gblocks, 128, 0, stream>>>(a, Wlin, out, nullptr, blin, N);
}